// MultiSampleContrastiveLoss_9637906612336
// MI455X (gfx1250) — compile-verified
//
#include <hip/hip_runtime.h>

typedef __attribute__((ext_vector_type(2))) float v2f;
typedef __attribute__((ext_vector_type(8))) float v8f;

#define TEMP_INV 10.0f   // 1 / 0.1
#define EPS_     1e-8f
#define B_  256
#define P_  32
#define N_  512
#define D_  256
#define K_  (P_ + N_)    // 544
#define TILES_ (K_ / 16) // 34

__global__ __launch_bounds__(256) void cl_main(const float* __restrict__ anchors,
                                               const float* __restrict__ positives,
                                               const float* __restrict__ negatives,
                                               float* __restrict__ acc) {
    __shared__ float ancS[D_];   // anchor row for this batch
    __shared__ float simS[K_];   // sim logits
    __shared__ float red[256];   // block reduction scratch

    const int b    = blockIdx.x;
    const int tid  = threadIdx.x;
    const int lane = tid & 31;   // wave32
    const int wave = tid >> 5;   // 8 waves / block

    // ---- preload anchor into LDS, compute ||a||^2 via block reduction ----
    float av = anchors[(size_t)b * D_ + tid];
    ancS[tid] = av;
    red[tid]  = av * av;
    __syncthreads();
    for (int s = 128; s > 0; s >>= 1) {
        if (tid < s) red[tid] += red[tid + s];
        __syncthreads();
    }
    const float na = sqrtf(red[0]);
    __syncthreads();

    // WMMA f32 16x16x4 lane mapping:
    //  A: lanes 0-15 -> row=lane, K={0,1}; lanes 16-31 -> row=lane-16, K={2,3}
    const int r    = lane & 15;           // matrix row owned by this lane
    const int dsel = (lane >> 4) << 1;    // 0 or 2: which K-pair of the chunk

    for (int t = wave; t < TILES_; t += 8) {
        const int k = t * 16 + r;         // global sample row
        const float* rowp = (k < P_)
            ? (positives + ((size_t)b * P_ + k) * D_)
            : (negatives + ((size_t)b * N_ + (k - P_)) * D_);

        v8f  c   = {};                    // fp32 accumulator tile
        float nrm = 0.0f;                 // per-lane partial sum of squares

        for (int d0 = 0; d0 < D_; d0 += 4) {
            v2f a;                        // A: sample tile fragment
            a.x = rowp[d0 + dsel + 0];
            a.y = rowp[d0 + dsel + 1];
            v2f bb;                       // B: anchor replicated across all 16 columns
            bb.x = ancS[d0 + dsel + 0];
            bb.y = ancS[d0 + dsel + 1];
            // D = A(16x4) * B(4x16) + C ; every output column == row-dot vs anchor
            c = __builtin_amdgcn_wmma_f32_16x16x4_f32(
                    false, a, false, bb, (short)0, c, false, false);
            nrm = fmaf(a.x, a.x, nrm);
            nrm = fmaf(a.y, a.y, nrm);
        }

        // fold the two K-halves of each row's squared norm (lanes L and L^16)
        float nk2 = nrm + __shfl_xor(nrm, 16, 32);
        float inv = TEMP_INV / fmaxf(na * sqrtf(nk2), EPS_);

        // C/D layout: VGPR m = row m on lanes 0-15, row 8+m on lanes 16-31.
        // Lane m     owns nk2[row m]    and dot(row m)    in c[m]
        // Lane 24+m  owns nk2[row 8+m]  and dot(row 8+m)  in c[m]
        #pragma unroll
        for (int m = 0; m < 8; ++m) {
            if (lane == m)      simS[t * 16 + m]     = c[m] * inv;
            if (lane == 24 + m) simS[t * 16 + 8 + m] = c[m] * inv;
        }
    }
    __syncthreads();

    // ---- softmax over K=544 in LDS ----
    float lm = -3.0e38f;
    for (int kk = tid; kk < K_; kk += 256) lm = fmaxf(lm, simS[kk]);
    red[tid] = lm;
    __syncthreads();
    for (int s = 128; s > 0; s >>= 1) {
        if (tid < s) red[tid] = fmaxf(red[tid], red[tid + s]);
        __syncthreads();
    }
    const float mx = red[0];
    __syncthreads();

    float ls = 0.0f;
    for (int kk = tid; kk < K_; kk += 256) ls += __expf(simS[kk] - mx);
    red[tid] = ls;
    __syncthreads();
    for (int s = 128; s > 0; s >>= 1) {
        if (tid < s) red[tid] += red[tid + s];
        __syncthreads();
    }
    const float sumExp = red[0];
    __syncthreads();

    red[tid] = (tid < P_) ? simS[tid] : 0.0f;   // sum of positive logits
    __syncthreads();
    for (int s = 128; s > 0; s >>= 1) {
        if (tid < s) red[tid] += red[tid + s];
        __syncthreads();
    }

    if (tid == 0) {
        // sum_{p<P} logp[p] = sum sim[p] - P*(max + log(sumExp))
        float contrib = red[0] - (float)P_ * (mx + __logf(sumExp));
        atomicAdd(acc, contrib);   // global_atomic_add_f32
    }
}

__global__ void cl_init(float* acc) { *acc = 0.0f; }

__global__ void cl_final(const float* __restrict__ acc, float* __restrict__ out) {
    out[0] = -(*acc) / (float)(B_ * P_);
}

extern "C" void kernel_launch(void* const* d_in, const int* in_sizes, int n_in,
                              void* d_out, int out_size, void* d_ws, size_t ws_size,
                              hipStream_t stream) {
    const float* anchors   = (const float*)d_in[0];
    const float* positives = (const float*)d_in[1];
    const float* negatives = (const float*)d_in[2];
    float* acc = (float*)d_ws;

    cl_init <<<1,   1,   0, stream>>>(acc);
    cl_main <<<B_,  256, 0, stream>>>(anchors, positives, negatives, acc);
    cl_final<<<1,   1,   0, stream>>>(acc, (float*)d_out);
}